// SpectralAttentionTorch_66288525246580
// MI455X (gfx1250) — compile-verified
//
#include <hip/hip_runtime.h>
#include <hip/hip_bf16.h>

// MI455X / gfx1250, wave32. All GEMMs via V_WMMA_F32_16X16X4_F32 (f32 in/acc).
// LDS tiles stored element-major with K contiguous (row stride 36 dwords):
//  - fragments load as single conflict-free ds_load_b64
//  - A/Q tiles stage as straight b128 async-to-LDS copies (GLOBAL_LOAD_ASYNC_TO_LDS)

typedef __attribute__((ext_vector_type(2))) float v2f;
typedef __attribute__((ext_vector_type(4))) float v4f;
typedef __attribute__((ext_vector_type(8))) float v8f;
typedef __attribute__((ext_vector_type(4))) int v4i;

#define EPSF 1e-5f
#define LDK 36  // LDS row stride in dwords: 32 K + 4 pad (16B aligned, no conflicts)

#if defined(__has_builtin)
#if __has_builtin(__builtin_amdgcn_global_load_async_to_lds_b128) && \
    __has_builtin(__builtin_amdgcn_global_load_async_to_lds_b32)
#define HAVE_ASYNC_LDS 1
#endif
#endif
#ifndef HAVE_ASYNC_LDS
#define HAVE_ASYNC_LDS 0
#endif

#define GAS __attribute__((address_space(1)))
#define LAS __attribute__((address_space(3)))

__device__ __forceinline__ void cp_b128(const float* g, float* l) {
#if HAVE_ASYNC_LDS
  __builtin_amdgcn_global_load_async_to_lds_b128((GAS v4i*)g, (LAS v4i*)l, 0, 0);
#else
  *(v4f*)l = *(const v4f*)g;
#endif
}

__device__ __forceinline__ void cp_b32(const float* g, float* l) {
#if HAVE_ASYNC_LDS
  __builtin_amdgcn_global_load_async_to_lds_b32((GAS int*)g, (LAS int*)l, 0, 0);
#else
  *l = *g;
#endif
}

__device__ __forceinline__ void async_join() {
#if HAVE_ASYNC_LDS
#if __has_builtin(__builtin_amdgcn_s_wait_asynccnt)
  __builtin_amdgcn_s_wait_asynccnt(0);
#else
  asm volatile("s_wait_asynccnt 0x0" ::: "memory");
#endif
#endif
}

__device__ __forceinline__ v8f wmma4(v2f a, v2f b, v8f c) {
  return __builtin_amdgcn_wmma_f32_16x16x4_f32(false, a, false, b, (short)0, c,
                                               false, false);
}

__device__ __forceinline__ v2f ldsv2(const float* p) { return *(const v2f*)p; }

// ---------------------------------------------------------------------------
// C[M,N] = (A[M,K] @ W[K,N] + bias[N]) * scale (row-major).
// 256 thr = 8 waves; macro tile 128x64; wave tile 32x32; K-chunk 32.
// ---------------------------------------------------------------------------
__global__ __launch_bounds__(256) void gemm_bias_kernel(
    const float* __restrict__ A, const float* __restrict__ W,
    const float* __restrict__ bias, float* __restrict__ C,
    int M, int N, int K, float scale)
{
  __shared__ float As[128 * LDK];  // [m][k]
  __shared__ float Bs[64 * LDK];   // [n][k] (transposed W tile)

  const int tid = threadIdx.x;
  const int lane = tid & 31;
  const int wv = tid >> 5;
  const int m_lane = lane & 15;
  const int kb0 = (lane >> 4) * 2;
  const int m0 = blockIdx.y * 128;
  const int n0 = blockIdx.x * 64;
  const int wm = wv & 3, wn = wv >> 2;
  const int rA0 = wm * 32, rA1 = wm * 32 + 16;
  const int cB0 = wn * 32, cB1 = wn * 32 + 16;

  v8f c00 = {}, c01 = {}, c10 = {}, c11 = {};

  for (int k0 = 0; k0 < K; k0 += 32) {
    __syncthreads();
#pragma unroll
    for (int r = 0; r < 4; ++r) {            // A tile: 1024 x b128, no transpose
      int f4 = r * 256 + tid;
      int i = f4 >> 3, j4 = f4 & 7;
      cp_b128(&A[(size_t)(m0 + i) * K + k0 + j4 * 4], &As[i * LDK + j4 * 4]);
    }
#pragma unroll
    for (int r = 0; r < 8; ++r) {            // B tile: 32x64 transposed scatter
      int flat = r * 256 + tid;
      int j = flat >> 6, n = flat & 63;
      cp_b32(&W[(size_t)(k0 + j) * N + n0 + n], &Bs[n * LDK + j]);
    }
    async_join();
    __syncthreads();
#pragma unroll
    for (int kk = 0; kk < 32; kk += 4) {
      const int kb = kk + kb0;
      v2f a0 = ldsv2(&As[(rA0 + m_lane) * LDK + kb]);
      v2f a1 = ldsv2(&As[(rA1 + m_lane) * LDK + kb]);
      v2f b0 = ldsv2(&Bs[(cB0 + m_lane) * LDK + kb]);
      v2f b1 = ldsv2(&Bs[(cB1 + m_lane) * LDK + kb]);
      c00 = wmma4(a0, b0, c00);
      c01 = wmma4(a0, b1, c01);
      c10 = wmma4(a1, b0, c10);
      c11 = wmma4(a1, b1, c11);
    }
  }
  const float bv0 = bias[n0 + cB0 + m_lane];
  const float bv1 = bias[n0 + cB1 + m_lane];
  const int r_hi = (lane >> 4) * 8;
#pragma unroll
  for (int r = 0; r < 8; ++r) {
    int row0 = m0 + rA0 + r + r_hi;
    int row1 = m0 + rA1 + r + r_hi;
    C[(size_t)row0 * N + n0 + cB0 + m_lane] = (c00[r] + bv0) * scale;
    C[(size_t)row0 * N + n0 + cB1 + m_lane] = (c01[r] + bv1) * scale;
    C[(size_t)row1 * N + n0 + cB0 + m_lane] = (c10[r] + bv0) * scale;
    C[(size_t)row1 * N + n0 + cB1 + m_lane] = (c11[r] + bv1) * scale;
  }
}

// ---------------------------------------------------------------------------
// Causal Toeplitz conv: Y[l,c] = sum_{t<=l} filt[l-t] * U[t,c] per (b,head).
// A-fragments gathered from LDS filter with causal predicate.
// grid.z: bit4 = tensor (k/v), low 4 = (b,head). grid.y = L/128.
// ---------------------------------------------------------------------------
__global__ __launch_bounds__(256) void toeplitz_conv_kernel(
    const float* __restrict__ kin, const float* __restrict__ vin,
    const float* __restrict__ sb,   // spectral_basis [1024][8]
    float* __restrict__ kout, float* __restrict__ vout)
{
  __shared__ float filtS[1024];
  __shared__ float Us[64 * LDK];   // [n][k] transposed

  const int zi = blockIdx.z;
  const int tensor = zi >> 4;
  const int bh = zi & 15;
  const int b = bh >> 3, head = bh & 7;
  const float* U = tensor ? vin : kin;
  float* Y = tensor ? vout : kout;

  const int tid = threadIdx.x;
  const int lane = tid & 31;
  const int wv = tid >> 5;
  const int m_lane = lane & 15;
  const int kb0 = (lane >> 4) * 2;
  const int m0 = blockIdx.y * 128;
  const int wm = wv & 3, wn = wv >> 2;
  const int rA0 = wm * 32, rA1 = wm * 32 + 16;
  const int cB0 = wn * 32, cB1 = wn * 32 + 16;

  for (int l = tid; l < 1024; l += 256) filtS[l] = sb[l * 8 + head];

  v8f c00 = {}, c01 = {}, c10 = {}, c11 = {};
  const int nChunks = (m0 >> 5) + 4;  // causal: t < m0+128
  const int l0 = m0 + rA0 + m_lane;
  const int l1 = m0 + rA1 + m_lane;

  for (int c = 0; c < nChunks; ++c) {
    const int t0 = c * 32;
    __syncthreads();
#pragma unroll
    for (int r = 0; r < 8; ++r) {
      int flat = r * 256 + tid;
      int j = flat >> 6, n = flat & 63;
      cp_b32(&U[(size_t)(b * 1024 + t0 + j) * 512 + head * 64 + n],
             &Us[n * LDK + j]);
    }
    async_join();
    __syncthreads();
#pragma unroll
    for (int kk = 0; kk < 32; kk += 4) {
      const int kb = kk + kb0;
      const int t = t0 + kb;
      v2f a0, a1;
      a0.x = (l0 >= t)     ? filtS[l0 - t]     : 0.f;
      a0.y = (l0 >= t + 1) ? filtS[l0 - t - 1] : 0.f;
      a1.x = (l1 >= t)     ? filtS[l1 - t]     : 0.f;
      a1.y = (l1 >= t + 1) ? filtS[l1 - t - 1] : 0.f;
      v2f b0 = ldsv2(&Us[(cB0 + m_lane) * LDK + kb]);
      v2f b1 = ldsv2(&Us[(cB1 + m_lane) * LDK + kb]);
      c00 = wmma4(a0, b0, c00);
      c01 = wmma4(a0, b1, c01);
      c10 = wmma4(a1, b0, c10);
      c11 = wmma4(a1, b1, c11);
    }
  }
  const int r_hi = (lane >> 4) * 8;
#pragma unroll
  for (int r = 0; r < 8; ++r) {
    int row0 = m0 + rA0 + r + r_hi;
    int row1 = m0 + rA1 + r + r_hi;
    Y[(size_t)(b * 1024 + row0) * 512 + head * 64 + cB0 + m_lane] = c00[r];
    Y[(size_t)(b * 1024 + row0) * 512 + head * 64 + cB1 + m_lane] = c01[r];
    Y[(size_t)(b * 1024 + row1) * 512 + head * 64 + cB0 + m_lane] = c10[r];
    Y[(size_t)(b * 1024 + row1) * 512 + head * 64 + cB1 + m_lane] = c11[r];
  }
}

// ---------------------------------------------------------------------------
// Gates + scan weights. One block per (b,h).
// g_l = relu(vc_l^T Wg kc_l + b)^2 + eps; G = prefix(g);
// S_t = suffix(1/(G+eps)); w_t = g_t * S_t.
// ---------------------------------------------------------------------------
__global__ __launch_bounds__(256) void gates_kernel(
    const float* __restrict__ kc, const float* __restrict__ vc,
    const float* __restrict__ wg_w, const float* __restrict__ wg_b,
    float* __restrict__ wout)
{
  __shared__ float WgS[4096];
  __shared__ float g_s[1024];
  __shared__ float inv_s[1024];
  __shared__ float sums[256];

  const int bh = blockIdx.x;
  const int b = bh >> 3, head = bh & 7;
  const int t = threadIdx.x;

  for (int i = t; i < 4096; i += 256) WgS[i] = wg_w[i];
  __syncthreads();
  const float gbias = wg_b[0];

  float gv[4];
  for (int p = 0; p < 4; ++p) {
    const int l = t * 4 + p;
    const float* kcl = kc + (size_t)(b * 1024 + l) * 512 + head * 64;
    const float* vcl = vc + (size_t)(b * 1024 + l) * 512 + head * 64;
    float tmp[64];
#pragma unroll
    for (int n = 0; n < 64; ++n) tmp[n] = 0.f;
    for (int m = 0; m < 64; ++m) {
      const float vm = vcl[m];
      const float* wrow = &WgS[m * 64];
#pragma unroll
      for (int n = 0; n < 64; ++n) tmp[n] = fmaf(vm, wrow[n], tmp[n]);
    }
    float logit = gbias;
#pragma unroll
    for (int n = 0; n < 64; ++n) logit = fmaf(tmp[n], kcl[n], logit);
    const float rl = fmaxf(logit, 0.f);
    gv[p] = rl * rl + EPSF;
    g_s[l] = gv[p];
  }

  float tot = gv[0] + gv[1] + gv[2] + gv[3];
  sums[t] = tot;
  __syncthreads();
  for (int off = 1; off < 256; off <<= 1) {
    float add = (t >= off) ? sums[t - off] : 0.f;
    __syncthreads();
    sums[t] += add;
    __syncthreads();
  }
  float run = sums[t] - tot;
#pragma unroll
  for (int p = 0; p < 4; ++p) {
    run += gv[p];
    inv_s[t * 4 + p] = 1.f / (run + EPSF);
  }
  __syncthreads();

  float rv[4];
#pragma unroll
  for (int p = 0; p < 4; ++p) rv[p] = inv_s[1023 - (t * 4 + p)];
  float rtot = rv[0] + rv[1] + rv[2] + rv[3];
  __syncthreads();
  sums[t] = rtot;
  __syncthreads();
  for (int off = 1; off < 256; off <<= 1) {
    float add = (t >= off) ? sums[t - off] : 0.f;
    __syncthreads();
    sums[t] += add;
    __syncthreads();
  }
  float rrun = sums[t] - rtot;
#pragma unroll
  for (int p = 0; p < 4; ++p) {
    rrun += rv[p];
    const int l = 1023 - (t * 4 + p);
    wout[(size_t)bh * 1024 + l] = g_s[l] * rrun;
  }
}

// ---------------------------------------------------------------------------
// attn_sum[b,h] = (w .* Vc)^T @ Kc : [64,1024]x[1024,64]. One block per (b,h).
// 8 waves: wave tile 16(M) x 32(N).
// ---------------------------------------------------------------------------
__global__ __launch_bounds__(256) void attn_accum_kernel(
    const float* __restrict__ kc, const float* __restrict__ vc,
    const float* __restrict__ wbuf, float* __restrict__ attn)
{
  __shared__ float VwS[64 * LDK];  // [m][t_rel] = w_t * vc[t][m]
  __shared__ float KS[64 * LDK];   // [n][t_rel]

  const int bh = blockIdx.x;
  const int b = bh >> 3, head = bh & 7;
  const int tid = threadIdx.x;
  const int lane = tid & 31;
  const int wv = tid >> 5;
  const int m_lane = lane & 15;
  const int kb0 = (lane >> 4) * 2;
  const int wm = wv & 3, wn = wv >> 2;
  const int rA = wm * 16;
  const int cB0 = wn * 32, cB1 = wn * 32 + 16;

  v8f c0 = {}, c1 = {};

  for (int t0 = 0; t0 < 1024; t0 += 32) {
    __syncthreads();
#pragma unroll
    for (int r = 0; r < 8; ++r) {
      int flat = r * 256 + tid;
      int j = flat >> 6, n = flat & 63;
      const float wrow = wbuf[(size_t)bh * 1024 + t0 + j];
      const size_t base = (size_t)(b * 1024 + t0 + j) * 512 + head * 64;
      VwS[n * LDK + j] = vc[base + n] * wrow;
      cp_b32(&kc[base + n], &KS[n * LDK + j]);
    }
    async_join();
    __syncthreads();
#pragma unroll
    for (int kk = 0; kk < 32; kk += 4) {
      const int kb = kk + kb0;
      v2f a  = ldsv2(&VwS[(rA + m_lane) * LDK + kb]);
      v2f b0 = ldsv2(&KS[(cB0 + m_lane) * LDK + kb]);
      v2f b1 = ldsv2(&KS[(cB1 + m_lane) * LDK + kb]);
      c0 = wmma4(a, b0, c0);
      c1 = wmma4(a, b1, c1);
    }
  }
  const int r_hi = (lane >> 4) * 8;
#pragma unroll
  for (int r = 0; r < 8; ++r) {
    const int row = rA + r + r_hi;
    attn[(size_t)bh * 4096 + row * 64 + cB0 + m_lane] = c0[r];
    attn[(size_t)bh * 4096 + row * 64 + cB1 + m_lane] = c1[r];
  }
}

// ---------------------------------------------------------------------------
// ctxt = Q_h @ attn_sum ([128 rows per block] x 64), then row-normalize.
// grid: (L/128, B*H). LDS pool: attnS + (QS reused as CS).
// ---------------------------------------------------------------------------
__global__ __launch_bounds__(256) void ctxt_norm_kernel(
    const float* __restrict__ q, const float* __restrict__ attn,
    float* __restrict__ unit)
{
  __shared__ float smem[64 * LDK + 128 * 72];
  __shared__ float scaleS[128];
  float* attnS = smem;             // [n][k=d], stride LDK
  float* QS = smem + 64 * LDK;     // [m][k=d], stride LDK (4608 < 9216, fits)
  float* CS = smem + 64 * LDK;     // [128][72], reused after compute

  const int bh = blockIdx.y;
  const int b = bh >> 3, head = bh & 7;
  const int m0 = blockIdx.x * 128;
  const int tid = threadIdx.x;
  const int lane = tid & 31;
  const int wv = tid >> 5;
  const int m_lane = lane & 15;
  const int kb0 = (lane >> 4) * 2;
  const int wm = wv & 3, wn = wv >> 2;
  const int rA0 = wm * 32, rA1 = wm * 32 + 16;
  const int cB0 = wn * 32, cB1 = wn * 32 + 16;

  for (int i = tid; i < 4096; i += 256) {    // attn_sum transposed: [n][k]
    int k = i >> 6, n = i & 63;
    cp_b32(&attn[(size_t)bh * 4096 + i], &attnS[n * LDK + k]);
  }
#pragma unroll
  for (int r = 0; r < 8; ++r) {              // Q tile, straight b128 copies
    int f4 = r * 256 + tid;
    int i = f4 >> 4, j4 = f4 & 15;
    cp_b128(&q[(size_t)(b * 1024 + m0 + i) * 512 + head * 64 + j4 * 4],
            &QS[i * LDK + j4 * 4]);
  }
  async_join();
  __syncthreads();

  v8f c00 = {}, c01 = {}, c10 = {}, c11 = {};
#pragma unroll
  for (int kk = 0; kk < 64; kk += 4) {
    const int kb = kk + kb0;
    v2f a0 = ldsv2(&QS[(rA0 + m_lane) * LDK + kb]);
    v2f a1 = ldsv2(&QS[(rA1 + m_lane) * LDK + kb]);
    v2f b0 = ldsv2(&attnS[(cB0 + m_lane) * LDK + kb]);
    v2f b1 = ldsv2(&attnS[(cB1 + m_lane) * LDK + kb]);
    c00 = wmma4(a0, b0, c00);
    c01 = wmma4(a0, b1, c01);
    c10 = wmma4(a1, b0, c10);
    c11 = wmma4(a1, b1, c11);
  }
  __syncthreads();  // all waves done reading QS; safe to reuse as CS

  const int r_hi = (lane >> 4) * 8;
#pragma unroll
  for (int r = 0; r < 8; ++r) {
    int row0 = rA0 + r + r_hi;
    int row1 = rA1 + r + r_hi;
    CS[row0 * 72 + cB0 + m_lane] = c00[r];
    CS[row0 * 72 + cB1 + m_lane] = c01[r];
    CS[row1 * 72 + cB0 + m_lane] = c10[r];
    CS[row1 * 72 + cB1 + m_lane] = c11[r];
  }
  __syncthreads();

  if (tid < 128) {
    float ss = 0.f;
#pragma unroll
    for (int c = 0; c < 64; ++c) {
      float v = CS[tid * 72 + c];
      ss = fmaf(v, v, ss);
    }
    scaleS[tid] = 1.f / fmaxf(sqrtf(ss), EPSF);
  }
  __syncthreads();

#pragma unroll
  for (int r = 0; r < 32; ++r) {
    int flat = r * 256 + tid;
    int i = flat >> 6, j = flat & 63;
    unit[(size_t)(b * 1024 + m0 + i) * 512 + head * 64 + j] =
        CS[i * 72 + j] * scaleS[i];
  }
}

// ---------------------------------------------------------------------------
extern "C" void kernel_launch(void* const* d_in, const int* in_sizes, int n_in,
                              void* d_out, int out_size, void* d_ws, size_t ws_size,
                              hipStream_t stream) {
  (void)in_sizes; (void)n_in; (void)out_size; (void)ws_size;
  const float* x    = (const float*)d_in[0];
  const float* sb   = (const float*)d_in[1];
  const float* wq_w = (const float*)d_in[2];
  const float* wq_b = (const float*)d_in[3];
  const float* wk_w = (const float*)d_in[4];
  const float* wk_b = (const float*)d_in[5];
  const float* wv_w = (const float*)d_in[6];
  const float* wv_b = (const float*)d_in[7];
  const float* wo_w = (const float*)d_in[8];
  const float* wo_b = (const float*)d_in[9];
  const float* wg_w = (const float*)d_in[10];
  const float* wg_b = (const float*)d_in[11];
  float* out = (float*)d_out;

  float* ws = (float*)d_ws;
  const size_t SZ = (size_t)2 * 1024 * 512;
  float* q    = ws;
  float* kbuf = ws + SZ;
  float* vbuf = ws + 2 * SZ;
  float* kc   = ws + 3 * SZ;
  float* vc   = ws + 4 * SZ;
  float* un   = ws + 5 * SZ;
  float* wpos = ws + 6 * SZ;            // [B*H*L] = 16384
  float* attn = ws + 6 * SZ + 16384;    // [B*H*64*64] = 65536

  dim3 blk(256);
  dim3 gGemm(8, 16);

  gemm_bias_kernel<<<gGemm, blk, 0, stream>>>(x, wq_w, wq_b, q,    2048, 512, 512, 1.0f);
  gemm_bias_kernel<<<gGemm, blk, 0, stream>>>(x, wk_w, wk_b, kbuf, 2048, 512, 512, 0.125f);
  gemm_bias_kernel<<<gGemm, blk, 0, stream>>>(x, wv_w, wv_b, vbuf, 2048, 512, 512, 1.0f);

  toeplitz_conv_kernel<<<dim3(1, 8, 32), blk, 0, stream>>>(kbuf, vbuf, sb, kc, vc);

  gates_kernel<<<dim3(16), blk, 0, stream>>>(kc, vc, wg_w, wg_b, wpos);

  attn_accum_kernel<<<dim3(16), blk, 0, stream>>>(kc, vc, wpos, attn);

  ctxt_norm_kernel<<<dim3(8, 16), blk, 0, stream>>>(q, attn, un);

  gemm_bias_kernel<<<gGemm, blk, 0, stream>>>(un, wo_w, wo_b, out, 2048, 512, 512, 1.0f);
}